// RetNet_83777632076217
// MI455X (gfx1250) — compile-verified
//
#include <hip/hip_runtime.h>
#include <hip/hip_bf16.h>

#define B_  512
#define H_  2048
#define NH_ 32
#define HS_ 64
#define K_  2048
#define EPS_ 1e-6f

typedef __attribute__((ext_vector_type(2))) float v2f;
typedef __attribute__((ext_vector_type(4))) float v4f;
typedef __attribute__((ext_vector_type(8))) float v8f;

#define LD2(p) (*(const v2f*)(p))

// D = A(16x4 f32) * B(4x16 f32) + C(16x16 f32), full f32 precision.
static __device__ __forceinline__ v8f wmma4(v2f a, v2f b, v8f c) {
  return __builtin_amdgcn_wmma_f32_16x16x4_f32(
      /*neg_a=*/false, a, /*neg_b=*/false, b,
      /*c_mod=*/(short)0, c, /*reuse_a=*/false, /*reuse_b=*/false);
}

// C[z][m][n] = sum_k A[m][k] * W_z[n][k]   (A @ W^T)
// One wave per block, 32(M) x 64(N) tile: 8 accumulators, 2 A-frags x 4 B-frags
// per K-step (6 loads : 8 WMMAs). K-steps are software-pipelined with register
// double-buffering so loads for step s+1 are in flight while step s computes.
__global__ void __launch_bounds__(32) gemm_xwT_wmma(
    const float* __restrict__ A,
    const float* __restrict__ W0, const float* __restrict__ W1,
    const float* __restrict__ W2, const float* __restrict__ W3,
    float* __restrict__ Cout, int M, int N, int K)
{
  const float* W;
  switch (blockIdx.z) {
    default: W = W0; break;
    case 1:  W = W1; break;
    case 2:  W = W2; break;
    case 3:  W = W3; break;
  }
  float* C = Cout + (size_t)blockIdx.z * (size_t)M * (size_t)N;

  const int lane = (int)(threadIdx.x & 31u);
  const int r    = lane & 15;
  const int hi   = lane >> 4;
  const int koff = hi << 1;                 // K sub-offset per 16x16x4 layout
  const int tn   = blockIdx.x * 64;
  const int tm   = blockIdx.y * 32;

  const float* a0p = A + (size_t)(tm +  0 + r) * K + koff;
  const float* a1p = A + (size_t)(tm + 16 + r) * K + koff;
  const float* w0p = W + (size_t)(tn +  0 + r) * K + koff;
  const float* w1p = W + (size_t)(tn + 16 + r) * K + koff;
  const float* w2p = W + (size_t)(tn + 32 + r) * K + koff;
  const float* w3p = W + (size_t)(tn + 48 + r) * K + koff;

  v8f acc00 = {}, acc01 = {}, acc02 = {}, acc03 = {};
  v8f acc10 = {}, acc11 = {}, acc12 = {}, acc13 = {};

  // Prologue: buffer A <- K-step 0
  v2f aA0 = LD2(a0p), aA1 = LD2(a1p);
  v2f bA0 = LD2(w0p), bA1 = LD2(w1p), bA2 = LD2(w2p), bA3 = LD2(w3p);

  for (int kb = 0; kb < K_ - 8; kb += 8) {
    // Weight streams come from HBM: prefetch 2KB ahead.
    __builtin_prefetch(w0p + kb + 512, 0, 1);
    __builtin_prefetch(w1p + kb + 512, 0, 1);
    __builtin_prefetch(w2p + kb + 512, 0, 1);
    __builtin_prefetch(w3p + kb + 512, 0, 1);

    // Load buffer B <- step kb+4 (overlaps with compute of buffer A).
    v2f aB0 = LD2(a0p + kb + 4), aB1 = LD2(a1p + kb + 4);
    v2f bB0 = LD2(w0p + kb + 4), bB1 = LD2(w1p + kb + 4);
    v2f bB2 = LD2(w2p + kb + 4), bB3 = LD2(w3p + kb + 4);

    // Compute step kb from buffer A.
    acc00 = wmma4(aA0, bA0, acc00);  acc01 = wmma4(aA0, bA1, acc01);
    acc02 = wmma4(aA0, bA2, acc02);  acc03 = wmma4(aA0, bA3, acc03);
    acc10 = wmma4(aA1, bA0, acc10);  acc11 = wmma4(aA1, bA1, acc11);
    acc12 = wmma4(aA1, bA2, acc12);  acc13 = wmma4(aA1, bA3, acc13);

    // Load buffer A <- step kb+8 (overlaps with compute of buffer B).
    aA0 = LD2(a0p + kb + 8);  aA1 = LD2(a1p + kb + 8);
    bA0 = LD2(w0p + kb + 8);  bA1 = LD2(w1p + kb + 8);
    bA2 = LD2(w2p + kb + 8);  bA3 = LD2(w3p + kb + 8);

    // Compute step kb+4 from buffer B.
    acc00 = wmma4(aB0, bB0, acc00);  acc01 = wmma4(aB0, bB1, acc01);
    acc02 = wmma4(aB0, bB2, acc02);  acc03 = wmma4(aB0, bB3, acc03);
    acc10 = wmma4(aB1, bB0, acc10);  acc11 = wmma4(aB1, bB1, acc11);
    acc12 = wmma4(aB1, bB2, acc12);  acc13 = wmma4(aB1, bB3, acc13);
  }

  // Epilogue: buffer A holds step K-8; last step is K-4.
  {
    v2f aB0 = LD2(a0p + K_ - 4), aB1 = LD2(a1p + K_ - 4);
    v2f bB0 = LD2(w0p + K_ - 4), bB1 = LD2(w1p + K_ - 4);
    v2f bB2 = LD2(w2p + K_ - 4), bB3 = LD2(w3p + K_ - 4);

    acc00 = wmma4(aA0, bA0, acc00);  acc01 = wmma4(aA0, bA1, acc01);
    acc02 = wmma4(aA0, bA2, acc02);  acc03 = wmma4(aA0, bA3, acc03);
    acc10 = wmma4(aA1, bA0, acc10);  acc11 = wmma4(aA1, bA1, acc11);
    acc12 = wmma4(aA1, bA2, acc12);  acc13 = wmma4(aA1, bA3, acc13);

    acc00 = wmma4(aB0, bB0, acc00);  acc01 = wmma4(aB0, bB1, acc01);
    acc02 = wmma4(aB0, bB2, acc02);  acc03 = wmma4(aB0, bB3, acc03);
    acc10 = wmma4(aB1, bB0, acc10);  acc11 = wmma4(aB1, bB1, acc11);
    acc12 = wmma4(aB1, bB2, acc12);  acc13 = wmma4(aB1, bB3, acc13);
  }

  // C/D layout: VGPR v holds row (v + 8*hi), lane r is the column.
#pragma unroll
  for (int v = 0; v < 8; ++v) {
    const size_t row0 = (size_t)(tm +  0 + v + 8 * hi) * (size_t)N;
    const size_t row1 = (size_t)(tm + 16 + v + 8 * hi) * (size_t)N;
    C[row0 + tn +  0 + r] = acc00[v];
    C[row0 + tn + 16 + r] = acc01[v];
    C[row0 + tn + 32 + r] = acc02[v];
    C[row0 + tn + 48 + r] = acc03[v];
    C[row1 + tn +  0 + r] = acc10[v];
    C[row1 + tn + 16 + r] = acc11[v];
    C[row1 + tn + 32 + r] = acc12[v];
    C[row1 + tn + 48 + r] = acc13[v];
  }
}

// One block of 64 threads per (b, head). The 64x64 state tile is streamed with
// b128 loads/stores: thread t owns a 4-column group c4=(t&15)*4 on the row
// rotation h = (t>>4) + 4*j. Fuses RoPE, decay+outer-product state update,
// einsum readout, RMSNorm and SiLU gating.
__global__ void __launch_bounds__(64) fused_state(
    const float* __restrict__ qbuf, const float* __restrict__ kbuf,
    const float* __restrict__ vbuf, const float* __restrict__ gbuf,
    const float* __restrict__ decay, const float* __restrict__ angle,
    const int*   __restrict__ seq,   const float* __restrict__ scales,
    const float* __restrict__ state_in,
    float* __restrict__ out_seq, float* __restrict__ out_scales,
    float* __restrict__ out_state, float* __restrict__ ybuf)
{
  const int bn = blockIdx.x;        // b*NH + n
  const int b  = bn >> 5;
  const int n  = bn & 31;
  const int t  = (int)threadIdx.x;  // 0..63

  __shared__ float qr[HS_];
  __shared__ float kr[HS_];
  __shared__ v4f   red4[HS_];
  __shared__ float red[16];

  const size_t vec_base = (size_t)b * H_ + (size_t)n * HS_;
  const float* qh = qbuf + vec_base;
  const float* kh = kbuf + vec_base;

  // RoPE element t: rot[2i] = -x[2i+1], rot[2i+1] = x[2i]; k gets HS^-0.5.
  const float so = (float)seq[b];
  const float th = so * angle[t];
  float s, c;
  __sincosf(th, &s, &c);
  const float qrot = (t & 1) ?  qh[t - 1]            : -qh[t + 1];
  const float krot = ((t & 1) ? kh[t - 1] : -kh[t + 1]) * 0.125f;
  qr[t] = qh[t] * c + qrot * s;
  kr[t] = kh[t] * 0.125f * c + krot * s;
  __syncthreads();

  const float d   = decay[n];
  const float ns  = scales[bn] * d + 1.0f;
  const float inv = rsqrtf(ns);

  const int rsub = t >> 4;          // starting row in the 4-row rotation
  const int c4   = (t & 15) << 2;   // 4-column group
  const v4f vv = *(const v4f*)(vbuf + vec_base + c4);

  const float* st  = state_in  + (size_t)bn * (HS_ * HS_);
  float*       sto = out_state + (size_t)bn * (HS_ * HS_);

  v4f acc = {};
#pragma unroll 4
  for (int h = rsub; h < HS_; h += 4) {
    v4f sv  = *(const v4f*)(st + h * HS_ + c4);     // global_load_b128
    v4f nsv = sv * d + vv * kr[h];                  // new_state row chunk
    *(v4f*)(sto + h * HS_ + c4) = nsv;              // global_store_b128
    acc += nsv * qr[h];
  }
  red4[t] = acc;
  __syncthreads();

  // Fold the 4 row-rotation partials per column group; start RMS reduction.
  if (t < 16) {
    v4f sum = red4[t] + red4[t + 16] + red4[t + 32] + red4[t + 48];
    sum = sum * inv;                                // einsum w/ scaled_state
    red[t]  = sum[0] * sum[0] + sum[1] * sum[1] + sum[2] * sum[2] + sum[3] * sum[3];
    red4[t] = sum;                                  // keep scaled readout
  }
  __syncthreads();
  if (t == 0) {
    float tot = 0.0f;
#pragma unroll
    for (int i = 0; i < 16; ++i) tot += red[i];
    red[0] = rsqrtf(tot * (1.0f / (float)HS_) + EPS_);
  }
  __syncthreads();

  if (t < 16) {
    const float rms = red[0];
    v4f o  = red4[t] * rms;
    v4f g4 = *(const v4f*)(gbuf + vec_base + c4);
    v4f y;
#pragma unroll
    for (int i = 0; i < 4; ++i) {
      const float g = g4[i];
      y[i] = (g / (1.0f + __expf(-g))) * o[i];
    }
    *(v4f*)(ybuf + vec_base + c4) = y;
  }

  if (t == 0) out_scales[bn] = ns;
  if (t == 0 && n == 0) out_seq[b] = (float)(seq[b] + 1);
}

extern "C" void kernel_launch(void* const* d_in, const int* in_sizes, int n_in,
                              void* d_out, int out_size, void* d_ws, size_t ws_size,
                              hipStream_t stream) {
  (void)in_sizes; (void)n_in; (void)out_size; (void)ws_size;

  const float* x      = (const float*)d_in[0];
  const float* Wq     = (const float*)d_in[1];
  const float* Wk     = (const float*)d_in[2];
  const float* Wv     = (const float*)d_in[3];
  const float* Wg     = (const float*)d_in[4];
  const float* Wo     = (const float*)d_in[5];
  const float* decay  = (const float*)d_in[6];
  const float* angle  = (const float*)d_in[7];
  const int*   seq    = (const int*)  d_in[8];
  const float* scales = (const float*)d_in[9];
  const float* state  = (const float*)d_in[10];

  // Workspace: q | k | v | g | y  (each B*H f32)
  float* qbuf = (float*)d_ws;
  float* kbuf = qbuf + (size_t)B_ * H_;
  float* vbuf = kbuf + (size_t)B_ * H_;
  float* gbuf = vbuf + (size_t)B_ * H_;
  float* ybuf = gbuf + (size_t)B_ * H_;

  // Output layout: out(B*H) | seq+1(B) | new_scales(B*NH) | new_state(B*NH*HS*HS)
  float* out_main   = (float*)d_out;
  float* out_seq    = out_main + (size_t)B_ * H_;
  float* out_scales = out_seq + B_;
  float* out_state  = out_scales + (size_t)B_ * NH_;

  // 1) q,k,v,g = x @ {Wq,Wk,Wv,Wg}^T   (f32 WMMA)
  {
    dim3 grid(H_ / 64, B_ / 32, 4);
    gemm_xwT_wmma<<<grid, dim3(32), 0, stream>>>(
        x, Wq, Wk, Wv, Wg, qbuf, B_, H_, K_);
  }

  // 2) Fused RoPE + state update + readout + RMSNorm + SiLU gate.
  fused_state<<<dim3(B_ * NH_), dim3(HS_), 0, stream>>>(
      qbuf, kbuf, vbuf, gbuf, decay, angle, seq, scales, state,
      out_seq, out_scales, out_state, ybuf);

  // 3) out = y @ Wo^T   (f32 WMMA)
  {
    dim3 grid(H_ / 64, B_ / 32, 1);
    gemm_xwT_wmma<<<grid, dim3(32), 0, stream>>>(
        ybuf, Wo, Wo, Wo, Wo, out_main, B_, H_, K_);
  }
}